// MyModel_61933428411888
// MI455X (gfx1250) — compile-verified
//
#include <hip/hip_runtime.h>
#include <hip/hip_bf16.h>
#include <math.h>

typedef __attribute__((ext_vector_type(2))) float v2f;
typedef __attribute__((ext_vector_type(8))) float v8f;

// -------------------------------------------------------------------------
// Kernel 1: d_vec[row] = rsqrt(rowsum(A[row,:])). Memory-bound: one 64MB pass.
// Also initializes the output flag to 1.0 (kernel2 only clears it).
// -------------------------------------------------------------------------
__global__ void rowsum_rsqrt_kernel(const float* __restrict__ A,
                                    float* __restrict__ dv,
                                    float* __restrict__ out,
                                    int N) {
    __shared__ float red[256];
    const int row = blockIdx.x;
    const int tid = threadIdx.x;

    const float4* rowp = (const float4*)(A + (size_t)row * N);
    const int nvec = N >> 2;           // 1024 float4 per row
    float s = 0.0f;
    for (int i = tid; i < nvec; i += blockDim.x) {
        float4 v = rowp[i];
        s += (v.x + v.y) + (v.z + v.w);
    }
    red[tid] = s;
    __syncthreads();
    for (int off = 128; off > 0; off >>= 1) {
        if (tid < off) red[tid] += red[tid + off];
        __syncthreads();
    }
    if (tid == 0) {
        dv[row] = rsqrtf(red[0]);
        if (row == 0) out[0] = 1.0f;   // optimistic init; verify pass clears on mismatch
    }
}

// -------------------------------------------------------------------------
// Kernel 2: one wave32 per 16x16 output tile.
//   dense tile  = WMMA( D_diag_block(16x16, built from d_vec), A tile )
//                 done as 4 x v_wmma_f32_16x16x4_f32 (K = 4 each)
//   sparse tile = d_vec[i] * A[i,j]  (VALU)
//   allclose compare -> clear flag on mismatch.
// Skipping the off-diagonal K-blocks of D is bit-exact (all terms are
// exact FP zeros), so this IS the reference dense GEMM on the nonzero work.
// -------------------------------------------------------------------------
__global__ void wmma_verify_kernel(const float* __restrict__ A,
                                   const float* __restrict__ dv,
                                   float* __restrict__ out,
                                   int N) {
    const int lane        = threadIdx.x & 31;
    const int waveInBlock = threadIdx.x >> 5;
    const int wavesPerBlk = blockDim.x >> 5;
    const int tileId      = blockIdx.x * wavesPerBlk + waveInBlock;
    const int tilesPerRow = N >> 4;

    const int ti = tileId / tilesPerRow;
    const int tj = tileId - ti * tilesPerRow;
    const int i0 = ti << 4;
    const int j0 = tj << 4;

    const int half = lane >> 4;    // 0: lanes 0-15, 1: lanes 16-31
    const int m    = lane & 15;

    // d_vec for this tile's rows (broadcast across the two lane halves)
    const float dvm = dv[i0 + m];

    v8f c = {};
#pragma unroll
    for (int t = 0; t < 4; ++t) {
        const int k0 = i0 + 4 * t;          // diagonal K-block slice

        // A-operand: D[i0+m, k0 .. k0+3], ISA 16x4 f32 layout:
        //   lanes 0-15:  vgpr0->K=k0+0, vgpr1->K=k0+1
        //   lanes 16-31: vgpr0->K=k0+2, vgpr1->K=k0+3
        // D[i,k] = (i==k) ? d_vec[i] : 0
        const int kRel = 4 * t + 2 * half;  // relative K of a.x within 0..15
        v2f a;
        a.x = (m == kRel)     ? dvm : 0.0f;
        a.y = (m == kRel + 1) ? dvm : 0.0f;

        // B-operand: A[k0 .. k0+3, j0 .. j0+15], 4x16 f32 layout:
        //   lanes 0-15:  vgpr0->K=k0+0, vgpr1->K=k0+1 (N = lane)
        //   lanes 16-31: vgpr0->K=k0+2, vgpr1->K=k0+3 (N = lane-16)
        const float* brow = A + (size_t)(k0 + 2 * half) * N + (j0 + m);
        v2f b;
        b.x = brow[0];
        b.y = brow[N];

        // D = A x B + C  (dense path, matrix pipe)
        c = __builtin_amdgcn_wmma_f32_16x16x4_f32(
                /*neg_a=*/false, a, /*neg_b=*/false, b,
                /*c_mod=*/(short)0, c, /*reuse_a=*/false, /*reuse_b=*/false);
    }

    // Sparse path + allclose compare (rtol=1e-5, atol=1e-8, like jnp.allclose).
    // C layout: vgpr r -> row M=r (lanes 0-15) / M=r+8 (lanes 16-31), col N=lane%16.
    int ok = 1;
#pragma unroll
    for (int r = 0; r < 8; ++r) {
        const int row = i0 + r + 8 * half;
        const int col = j0 + m;
        const float sparse = dv[row] * A[(size_t)row * N + col];
        const float dense  = c[r];
        const float diff   = fabsf(sparse - dense);
        const float tol    = 1e-8f + 1e-5f * fabsf(dense);
        if (diff > tol) ok = 0;
    }
    if (!ok) out[0] = 0.0f;   // all writers write the same value; race-free result
}

// -------------------------------------------------------------------------
extern "C" void kernel_launch(void* const* d_in, const int* in_sizes, int n_in,
                              void* d_out, int out_size, void* d_ws, size_t ws_size,
                              hipStream_t stream) {
    const float* A   = (const float*)d_in[0];
    float*       out = (float*)d_out;
    float*       dv  = (float*)d_ws;    // N floats of scratch

    // N*N = in_sizes[0]  (N = 4096 for this problem)
    int N = (int)lroundf(sqrtf((float)in_sizes[0]));

    // Pass 1: row sums -> d_vec = rsqrt(deg); also sets out[0] = 1.0
    rowsum_rsqrt_kernel<<<N, 256, 0, stream>>>(A, dv, out, N);

    // Pass 2: WMMA dense path vs sparse path, one wave per 16x16 tile.
    const int tiles         = (N >> 4) * (N >> 4);  // 65536
    const int wavesPerBlock = 8;                    // 256 threads = 8 wave32
    const int blocks        = tiles / wavesPerBlock;
    wmma_verify_kernel<<<blocks, 256, 0, stream>>>(A, dv, out, N);
}